// GCN_27711128994646
// MI455X (gfx1250) — compile-verified
//
#include <hip/hip_runtime.h>

// ---------------------------------------------------------------------------
// 3-layer GCN (DGL GraphConv, norm='both') for MI455X / gfx1250.
// GEMMs use V_WMMA_F32_16X16X4_F32 (full fp32 precision, fp32 accumulate).
// Scatter-adds use global_atomic_add_f32.  wave32 / one wave per WMMA tile.
// ---------------------------------------------------------------------------

typedef __attribute__((ext_vector_type(2))) float v2f;
typedef __attribute__((ext_vector_type(8))) float v8f;

#define NNODES 100000

// ----------------------------- utility kernels -----------------------------

__global__ void zero_f32(float* __restrict__ p, int n) {
  int i = blockIdx.x * blockDim.x + threadIdx.x;
  if (i < n) p[i] = 0.0f;
}

__global__ void degree_kernel(const int* __restrict__ src, const int* __restrict__ dst,
                              float* __restrict__ deg_out, float* __restrict__ deg_in, int E) {
  int e = blockIdx.x * blockDim.x + threadIdx.x;
  if (e < E) {
    atomicAdd(&deg_out[src[e]], 1.0f);
    atomicAdd(&deg_in[dst[e]], 1.0f);
  }
}

// in-place: deg -> clip(deg,1)^-0.5
__global__ void norm_kernel(float* __restrict__ deg_out, float* __restrict__ deg_in, int n) {
  int i = blockIdx.x * blockDim.x + threadIdx.x;
  if (i < n) {
    deg_out[i] = rsqrtf(fmaxf(deg_out[i], 1.0f));
    deg_in[i]  = rsqrtf(fmaxf(deg_in[i], 1.0f));
  }
}

// agg[dst[e]*F + f] += x[src[e]*F + f] * (SCALE ? scale[src[e]] : 1)
template <int F, bool SCALE>
__global__ void scatter_kernel(const float* __restrict__ x,
                               const int* __restrict__ src, const int* __restrict__ dst,
                               const float* __restrict__ scale,
                               float* __restrict__ agg, int E) {
  int tid = blockIdx.x * blockDim.x + threadIdx.x;   // E*F <= 102.4M < 2^31
  int e = tid / F;
  int f = tid - e * F;
  if (e < E) {
    int s = src[e];
    int d = dst[e];
    float v = x[s * F + f];
    if (SCALE) v *= scale[s];
    atomicAdd(&agg[d * F + f], v);
  }
}

// out[i] = (RELU? relu : id)(agg[i]*invIn[row] + bias[col]), i over N*F
template <int F, bool RELU>
__global__ void epilogue_kernel(const float* __restrict__ agg,
                                const float* __restrict__ invIn,
                                const float* __restrict__ bias,
                                float* __restrict__ out, int n) {
  int i = blockIdx.x * blockDim.x + threadIdx.x;
  if (i < n) {
    int row = i / F;
    int f = i - row * F;
    float v = agg[i] * invIn[row] + bias[f];
    if (RELU) v = fmaxf(v, 0.0f);
    out[i] = v;
  }
}

// ------------------------------- WMMA GEMM ---------------------------------
// One wave (32 lanes) per 16x16 output tile; K accumulated 4 at a time with
// V_WMMA_F32_16X16X4_F32.  A is MxK row-major, B is KxNC row-major, D is MxNC.
// Optional: prescale A rows (inv_out), fused epilogue (inv_in*.. + bias, relu).
// M % 16 == 0 always (100000 = 6250*16); only NC=40 needs a column guard,
// implemented with selects (no EXEC divergence — WMMA needs EXEC all-ones).
template <int K, int NC, bool PRESCALE, bool EPI, bool RELU>
__global__ __launch_bounds__(32) void gemm_wmma(const float* __restrict__ A,
                                                const float* __restrict__ B,
                                                float* __restrict__ D,
                                                const float* __restrict__ rowScaleA,
                                                const float* __restrict__ rowScaleOut,
                                                const float* __restrict__ bias) {
  const int lane = threadIdx.x;      // 0..31, wave32
  const int half = lane >> 4;        // 0: lanes 0-15, 1: lanes 16-31
  const int l15  = lane & 15;
  const int tileM = blockIdx.x * 16;
  const int tileN = blockIdx.y * 16;

  const int arow = tileM + l15;                  // A row this lane reads
  const float* __restrict__ Ap = A + arow * K;
  float ascale = 1.0f;
  if (PRESCALE) ascale = rowScaleA[arow];

  const int  col   = tileN + l15;                // B/D column this lane touches
  const bool colOK = (NC % 16 == 0) ? true : (col < NC);

  v8f acc = {0.f, 0.f, 0.f, 0.f, 0.f, 0.f, 0.f, 0.f};

#pragma unroll 4
  for (int k = 0; k < K; k += 4) {
    const int ka = k + half * 2;                 // lanes 0-15: K=k,k+1 ; 16-31: K=k+2,k+3
    v2f a, b;
    a.x = Ap[ka];
    a.y = Ap[ka + 1];
    if (PRESCALE) { a.x *= ascale; a.y *= ascale; }
    b.x = colOK ? B[ka * NC + col] : 0.0f;
    b.y = colOK ? B[(ka + 1) * NC + col] : 0.0f;
    // 8 args: (neg_a, A, neg_b, B, c_mod, C, reuse_a, reuse_b)
    acc = __builtin_amdgcn_wmma_f32_16x16x4_f32(false, a, false, b, (short)0, acc,
                                                false, false);
  }

  if (colOK) {
    const int mbase = tileM + half * 8;          // lanes 0-15: M=r ; lanes 16-31: M=8+r
#pragma unroll
    for (int r = 0; r < 8; ++r) {
      const int row = mbase + r;
      float v = acc[r];
      if (EPI)  v = v * rowScaleOut[row] + bias[col];
      if (RELU) v = fmaxf(v, 0.0f);
      D[row * NC + col] = v;
    }
  }
}

// ------------------------------- launcher ----------------------------------

extern "C" void kernel_launch(void* const* d_in, const int* in_sizes, int n_in,
                              void* d_out, int out_size, void* d_ws, size_t ws_size,
                              hipStream_t stream) {
  const float* features = (const float*)d_in[0];
  const int*   src      = (const int*)d_in[1];
  const int*   dst      = (const int*)d_in[2];
  const float* W1       = (const float*)d_in[3];
  const float* b1       = (const float*)d_in[4];
  const float* W2       = (const float*)d_in[5];
  const float* b2       = (const float*)d_in[6];
  const float* W3       = (const float*)d_in[7];
  const float* b3       = (const float*)d_in[8];

  const int N = NNODES;
  const int E = in_sizes[1];

  // workspace layout (floats): inv_out[N] inv_in[N] P0[N*256] P1[N*128] P2[N*128]
  float* ws      = (float*)d_ws;
  float* inv_out = ws;
  float* inv_in  = ws + N;
  float* P0      = ws + 2 * (size_t)N;
  float* P1      = P0 + (size_t)N * 256;
  float* P2      = P1 + (size_t)N * 128;
  float* H3      = P0;                       // N*40, P0 dead after GEMM2
  float* AGG3    = P0 + (size_t)N * 40;      // N*40

  const int T = 256;
  auto blocks = [T](long long n) { return (unsigned)((n + T - 1) / T); };

  // degree norms
  zero_f32<<<blocks(2LL * N), T, 0, stream>>>(inv_out, 2 * N);   // zeros both deg arrays
  degree_kernel<<<blocks(E), T, 0, stream>>>(src, dst, inv_out, inv_in, E);
  norm_kernel<<<blocks(N), T, 0, stream>>>(inv_out, inv_in, N);

  // ---- layer 1: aggregate first (128 -> 256) ----
  zero_f32<<<blocks((long long)N * 128), T, 0, stream>>>(P1, N * 128);
  scatter_kernel<128, true><<<blocks((long long)E * 128), T, 0, stream>>>(
      features, src, dst, inv_out, P1, E);
  gemm_wmma<128, 256, false, true, true><<<dim3(N / 16, 256 / 16), 32, 0, stream>>>(
      P1, W1, P0, nullptr, inv_in, b1);                          // out1 = P0 (relu fused)

  // ---- layer 2: transform first (256 -> 128) ----
  gemm_wmma<256, 128, true, false, false><<<dim3(N / 16, 128 / 16), 32, 0, stream>>>(
      P0, W2, P2, inv_out, nullptr, nullptr);                    // h2b = P2
  zero_f32<<<blocks((long long)N * 128), T, 0, stream>>>(P1, N * 128);
  scatter_kernel<128, false><<<blocks((long long)E * 128), T, 0, stream>>>(
      P2, src, dst, nullptr, P1, E);                             // agg2 = P1
  epilogue_kernel<128, true><<<blocks((long long)N * 128), T, 0, stream>>>(
      P1, inv_in, b2, P2, N * 128);                              // out2 = P2 (relu)

  // ---- layer 3: transform first (128 -> 40) ----
  gemm_wmma<128, 40, true, false, false><<<dim3(N / 16, 3), 32, 0, stream>>>(
      P2, W3, H3, inv_out, nullptr, nullptr);                    // h3b = H3
  zero_f32<<<blocks((long long)N * 40), T, 0, stream>>>(AGG3, N * 40);
  scatter_kernel<40, false><<<blocks((long long)E * 40), T, 0, stream>>>(
      H3, src, dst, nullptr, AGG3, E);
  epilogue_kernel<40, false><<<blocks((long long)N * 40), T, 0, stream>>>(
      AGG3, inv_in, b3, (float*)d_out, N * 40);                  // final logits
}